// SoftBoundDDM_RT_36112085025059
// MI455X (gfx1250) — compile-verified
//
#include <hip/hip_runtime.h>
#include <stdint.h>
#include <math.h>

// ---------------------------------------------------------------------------
// SoftBound DDM: e[t] = decay*e[t-1] + u[t-1], dv/h1/h0 pointwise.
// Strategy: chunked parallel linear recurrence (C=16 chunks of L=128 steps)
//   K1: per-(trial,chunk) partial sums S_c = sum decay^(L-1-k) u[cL+k]
//   K2: per-trial 16-step carry scan  e[cL]
//   K3: per-(trial,chunk) emission of dv/h1/h0
// Data movement: CDNA5 global_load_async_to_lds_b128 stages 64x32 tiles into
// LDS (coalesced global reads, per-lane LDS addresses = transpose for free),
// double-buffered with s_wait_asynccnt.
// ---------------------------------------------------------------------------

#define NTR   4096      // trials
#define NT    2000      // timesteps
#define CHK   128       // chunk length (timesteps per chunk)
#define NCH   16        // number of chunks (16*128 = 2048 >= 2000, last ragged)
#define NPART 15        // partial sums needed (chunks 0..14 feed carries 1..15)
#define TPB   64        // threads per block == trials per block (2 waves)
#define STRW  36        // LDS row stride in words (32 cols + 4 pad, 16B aligned)
#define STRB  (STRW * 4)
#define PLW   (TPB * STRW)      // words per (buffer, array) plane

__device__ __forceinline__ unsigned lds_lo32(const void* p) {
  // Generic pointers to LDS carry the LDS byte offset in the low 32 bits
  // async-to-LDS VDST wants that offset.
  return (unsigned)(uintptr_t)p;
}

// Stage one 64-trial x 32-timestep tile of stim and noise into LDS planes.
// Global side: 128B-coalesced (8 lanes x 16B cover one 128B row segment).
// LDS side: row-major [trial][col] with stride 36 words.
// Each wave issues exactly 16 async instructions per call.
__device__ __forceinline__ void stage_tiles(const float* __restrict__ stim,
                                            const float* __restrict__ noise,
                                            unsigned lds_s, unsigned lds_n,
                                            int T0, int col0, int tid,
                                            bool need_guard) {
#pragma unroll
  for (int it = 0; it < 8; ++it) {
    int flat = it * TPB + tid;   // 0..511
    int row  = flat >> 3;        // 0..63  (trial within block)
    int cg   = flat & 7;         // 16B column group (4 floats)
    int colE = col0 + cg * 4;    // first time index of this group
    unsigned voff = (unsigned)(((T0 + row) * NT + colE) * 4);
    unsigned ls   = lds_s + (unsigned)(row * STRB + cg * 16);
    unsigned ln   = lds_n + (unsigned)(row * STRB + cg * 16);
    bool ok = (!need_guard) || (colE < NT);   // whole 4-float group in-bounds
    if (ok) {
      asm volatile("global_load_async_to_lds_b128 %0, %1, %2"
                   :: "v"(ls), "v"(voff), "s"((uint64_t)(uintptr_t)stim)
                   : "memory");
      asm volatile("global_load_async_to_lds_b128 %0, %1, %2"
                   :: "v"(ln), "v"(voff), "s"((uint64_t)(uintptr_t)noise)
                   : "memory");
    }
  }
}

__device__ __forceinline__ void wait_stage(int s) {
  // 16 asyncs per stage per wave; in-order completion => <=16 leaves only the
  // newest (prefetched) stage outstanding. Last stage: drain fully.
  if (s < 3) asm volatile("s_wait_asynccnt 16" ::: "memory");
  else       asm volatile("s_wait_asynccnt 0"  ::: "memory");
}

// ---------------------------------------------------------------------------
// K1: partial sums. ssum = sum_{k=0}^{127} decay^(127-k) u[cL+k]
// ---------------------------------------------------------------------------
__global__ void __launch_bounds__(TPB)
ddm_partials(const float* __restrict__ stim, const float* __restrict__ noise,
             const float* __restrict__ a_p, const float* __restrict__ z_p,
             const float* __restrict__ g_p, const float* __restrict__ o_p,
             float* __restrict__ S) {
  __shared__ float smem[2][2][PLW];
  const int tid = threadIdx.x;
  const int T0  = blockIdx.x * TPB;
  const int c   = blockIdx.y;            // 0..14, all columns < 1792+128 <= 2000
  const int cb  = c * CHK;

  const float a = *a_p, z = *z_p, gain = *g_p, off = *o_p;
  const float sp   = z * a;
  const float sqdt = sqrtf(0.001f);
  const float gdt  = gain * 0.001f;
  const float addc = off * 0.001f + 0.01f * sp * 0.001f;

  const unsigned lb = lds_lo32(&smem[0][0][0]);
  const unsigned PB = (unsigned)(PLW * 4);

  stage_tiles(stim, noise, lb + 0u * PB, lb + 1u * PB, T0, cb +  0, tid, false);
  stage_tiles(stim, noise, lb + 2u * PB, lb + 3u * PB, T0, cb + 32, tid, false);

  float ssum = 0.f;
  for (int s = 0; s < 4; ++s) {
    wait_stage(s);
    __syncthreads();
    const float* st = &smem[s & 1][0][tid * STRW];
    const float* no = &smem[s & 1][1][tid * STRW];
#pragma unroll
    for (int j = 0; j < 32; ++j) {
      float u = fmaf(gdt, st[j], fmaf(sqdt, no[j], addc));
      ssum = fmaf(0.99999f, ssum, u);
    }
    __syncthreads();
    if (s + 2 < 4)
      stage_tiles(stim, noise,
                  lb + (unsigned)((s & 1) * 2) * PB,
                  lb + (unsigned)((s & 1) * 2 + 1) * PB,
                  T0, cb + (s + 2) * 32, tid, false);
  }
  S[c * NTR + T0 + tid] = ssum;
}

// ---------------------------------------------------------------------------
// K2: carry scan. carries[c] = e[c*128]; e[c] = decay^128 * e[c-1] + S[c-1]
// ---------------------------------------------------------------------------
__global__ void __launch_bounds__(256)
ddm_scan(const float* __restrict__ S,
         const float* __restrict__ a_p, const float* __restrict__ z_p,
         float* __restrict__ carries) {
  const int trial = blockIdx.x * blockDim.x + threadIdx.x;
  if (trial >= NTR) return;
  const float sp = (*z_p) * (*a_p);
  float dL = 0.99999f;
#pragma unroll
  for (int i = 0; i < 7; ++i) dL *= dL;   // decay^128
  float e = sp;
  carries[trial] = e;
#pragma unroll
  for (int c = 1; c < NCH; ++c) {
    e = fmaf(dL, e, S[(c - 1) * NTR + trial]);
    carries[c * NTR + trial] = e;
  }
}

// ---------------------------------------------------------------------------
// K3: emission. Given e[c*128], run 128 steps, produce dv/h1/h0.
// ---------------------------------------------------------------------------
__global__ void __launch_bounds__(TPB)
ddm_emit(const float* __restrict__ stim, const float* __restrict__ noise,
         const float* __restrict__ a_p, const float* __restrict__ z_p,
         const float* __restrict__ g_p, const float* __restrict__ o_p,
         const float* __restrict__ b_p,
         const float* __restrict__ carries, float* __restrict__ out) {
  __shared__ float smem[2][2][PLW];
  const int tid   = threadIdx.x;
  const int T0    = blockIdx.x * TPB;
  const int c     = blockIdx.y;          // 0..15 (chunk 15 ragged: t<2000)
  const int cb    = c * CHK;
  const int trial = T0 + tid;

  const float a = *a_p, z = *z_p, gain = *g_p, off = *o_p, beta = *b_p;
  const float sp   = z * a;
  const float sqdt = sqrtf(0.001f);
  const float gdt  = gain * 0.001f;
  const float addc = off * 0.001f + 0.01f * sp * 0.001f;

  float e = carries[c * NTR + trial];

  float* __restrict__ dv_o = out;
  float* __restrict__ h1_o = out + (size_t)NTR * NT;
  float* __restrict__ h0_o = out + 2 * (size_t)NTR * NT;

  const unsigned lb = lds_lo32(&smem[0][0][0]);
  const unsigned PB = (unsigned)(PLW * 4);
  const bool guard = (c == NCH - 1);

  stage_tiles(stim, noise, lb + 0u * PB, lb + 1u * PB, T0, cb +  0, tid, guard);
  stage_tiles(stim, noise, lb + 2u * PB, lb + 3u * PB, T0, cb + 32, tid, guard);

  float u_carry = 0.f;   // u at the last column of the previous sub-tile
  for (int s = 0; s < 4; ++s) {
    wait_stage(s);
    __syncthreads();
    const float* st = &smem[s & 1][0][tid * STRW];
    const float* no = &smem[s & 1][1][tid * STRW];
#pragma unroll
    for (int g = 0; g < 8; ++g) {
      const int j0 = s * 32 + g * 4;     // step within chunk
      const int t0 = cb + j0;            // global time of group start (mult of 4)
      if (t0 < NT) {                     // whole 4-group valid (2000 % 4 == 0)
        float dva[4], h1a[4], h0a[4];
#pragma unroll
        for (int q = 0; q < 4; ++q) {
          const int j  = j0 + q;
          const int lc = g * 4 + q;      // j & 31 (s*32 drops out)
          if (j > 0) {
            float u = (lc == 0) ? u_carry
                    : fmaf(gdt, st[lc - 1], fmaf(sqdt, no[lc - 1], addc));
            e = fmaf(0.99999f, e, u);
          }
          const float tf = (float)(t0 + q);
          const float dv = fmaf(tf * 0.01f, e - sp, sp);
          const float x1 = __expf(-beta * (dv - a));
          const float x0 = __expf(beta * dv);
          dva[q] = dv;
          h1a[q] = __fdividef(1.f, 1.f + x1);
          h0a[q] = __fdividef(1.f, 1.f + x0);
        }
        const size_t o = (size_t)trial * NT + t0;
        *(float4*)(dv_o + o) = make_float4(dva[0], dva[1], dva[2], dva[3]);
        *(float4*)(h1_o + o) = make_float4(h1a[0], h1a[1], h1a[2], h1a[3]);
        *(float4*)(h0_o + o) = make_float4(h0a[0], h0a[1], h0a[2], h0a[3]);
      }
    }
    // u at column 31 carries into the next sub-tile (garbage only when the
    // next sub-tile is fully past t=2000, in which case it is never consumed).
    u_carry = fmaf(gdt, st[31], fmaf(sqdt, no[31], addc));
    __syncthreads();
    if (s + 2 < 4)
      stage_tiles(stim, noise,
                  lb + (unsigned)((s & 1) * 2) * PB,
                  lb + (unsigned)((s & 1) * 2 + 1) * PB,
                  T0, cb + (s + 2) * 32, tid, guard);
  }
}

// ---------------------------------------------------------------------------
extern "C" void kernel_launch(void* const* d_in, const int* in_sizes, int n_in,
                              void* d_out, int out_size, void* d_ws, size_t ws_size,
                              hipStream_t stream) {
  (void)in_sizes; (void)n_in; (void)out_size; (void)ws_size;
  const float* stim  = (const float*)d_in[0];
  const float* noise = (const float*)d_in[1];
  const float* a_p   = (const float*)d_in[2];
  const float* z_p   = (const float*)d_in[3];
  const float* g_p   = (const float*)d_in[4];
  const float* o_p   = (const float*)d_in[5];
  const float* b_p   = (const float*)d_in[6];
  float* out = (float*)d_out;

  float* S       = (float*)d_ws;          // NPART * NTR floats
  float* carries = S + NPART * NTR;       // NCH   * NTR floats

  ddm_partials<<<dim3(NTR / TPB, NPART), TPB, 0, stream>>>(
      stim, noise, a_p, z_p, g_p, o_p, S);
  ddm_scan<<<NTR / 256, 256, 0, stream>>>(S, a_p, z_p, carries);
  ddm_emit<<<dim3(NTR / TPB, NCH), TPB, 0, stream>>>(
      stim, noise, a_p, z_p, g_p, o_p, b_p, carries, out);
}